// GraphTokenPool_89154931130440
// MI455X (gfx1250) — compile-verified
//
#include <hip/hip_runtime.h>

// Problem constants (from reference)
#define Bsz   32
#define Cdim  384
#define Ndim  4096
#define Mdim  64
#define Rdim  16
#define NT    64          // N-columns per tile
#define TILES_PER_WG 8    // consecutive N-tiles per workgroup (T-accumulator persistence)
#define WGS_PER_BATCH 8   // 64 tiles / 8

// LDS strides in floats (padded for bank-conflict-free strided fragment reads,
// and multiples of 4 so float4/TDM stores stay aligned)
#define XS  68    // sX  [384][68]  (64 data + 4 pad DWORDs per row -> TDM pad feature)
#define VWS 388   // sVw [16][388]
#define US  20    // sU  [64][20]
#define VS  66    // sV  [16][66]
#define SS  68    // sS  [64][68]

#define XBUF   (Cdim * XS)             // 26112 floats per X buffer
#define OFF_X0 0
#define OFF_X1 XBUF                    // 26112
#define OFF_VW (2 * XBUF)              // 52224
#define OFF_U  (OFF_VW + Rdim * VWS)   // 58432
#define OFF_V  (OFF_U  + Mdim * US)    // 59712
#define OFF_S  (OFF_V  + Rdim * VS)    // 60768
#define LDS_FLOATS (OFF_S + Mdim * SS) // 65120 floats = 260480 bytes

typedef __attribute__((ext_vector_type(2))) float v2f;
typedef __attribute__((ext_vector_type(8))) float v8f;
typedef __attribute__((ext_vector_type(4))) unsigned int u32x4;
typedef __attribute__((ext_vector_type(8))) int i32x8;
typedef __attribute__((ext_vector_type(4))) int i32x4;

__device__ __forceinline__ v8f wmma_f32(v2f a, v2f b, v8f c) {
  // D = A(16x4) * B(4x16) + C(16x16), f32, wave32
  return __builtin_amdgcn_wmma_f32_16x16x4_f32(false, a, false, b, (short)0, c,
                                               false, false);
}

// Tensor Data Mover: DMA a 64x384 f32 tile (row stride 4096 floats in memory)
// into LDS at lds_byte_off, inserting 4 pad DWORDs after every 64 data DWORDs
// so the LDS row stride becomes XS=68 floats (bank-conflict-free layout).
__device__ __forceinline__ void tdm_load_x_tile(const float* gsrc,
                                                unsigned lds_byte_off) {
  unsigned long long ga = (unsigned long long)(uintptr_t)gsrc;
  u32x4 g0;
  g0[0] = 1u;                                      // count=1, user descriptor
  g0[1] = lds_byte_off;                            // lds_addr [63:32]
  g0[2] = (unsigned)(ga & 0xFFFFFFFFu);            // global_addr [95:64]
  g0[3] = (unsigned)((ga >> 32) & 0x1FFFFFFu)      // global_addr [120:96]
          | (2u << 30);                            // type=2 ("image")
  i32x8 g1;
  g1[0] = (2 << 16)      // data_size = 4 bytes
        | (1 << 20)      // pad_enable
        | (5 << 22)      // pad_interval: every 64 DWORDs
        | (3 << 25);     // pad_amount: 4 DWORDs
  g1[1] = (int)(((unsigned)Ndim & 0xFFFFu) << 16);             // tensor_dim0 lo16
  g1[2] = (int)(((unsigned)Ndim >> 16) | ((unsigned)Cdim << 16)); // dim0 hi / dim1 lo
  g1[3] = (int)(((unsigned)Cdim >> 16) | ((unsigned)NT << 16));   // dim1 hi / tile_dim0
  g1[4] = (int)Cdim;                                           // tile_dim1 (tile_dim2=0)
  g1[5] = (int)Ndim;                                           // tensor_dim0_stride lo32
  g1[6] = 0;
  g1[7] = 0;
  i32x4 z4 = {0, 0, 0, 0};
  i32x8 z8 = {0, 0, 0, 0, 0, 0, 0, 0};
  __builtin_amdgcn_tensor_load_to_lds(g0, g1, z4, z4, z8, 0);
}

__global__ void zero_f32_kernel(float* __restrict__ p, int n) {
  int i = blockIdx.x * 256 + threadIdx.x;
  if (i < n) p[i] = 0.0f;
}

extern "C" __global__ void __launch_bounds__(256)
tokenpool_fused_kernel(const float* __restrict__ x, const float* __restrict__ Vw,
                       const float* __restrict__ Vb, const float* __restrict__ U,
                       float* __restrict__ Tout, float* __restrict__ Sout) {
  extern __shared__ float smem[];
  float* sVw = smem + OFF_VW;
  float* sU  = smem + OFF_U;
  float* sV  = smem + OFF_V;
  float* sS  = smem + OFF_S;

  const int tid  = threadIdx.x;
  const int wave = tid >> 5;
  const int lane = tid & 31;
  const int l16  = lane & 15;
  const int hf   = lane >> 4;          // lane half: 0 (lanes 0-15) / 1 (lanes 16-31)
  const int bidx = blockIdx.x >> 3;    // batch
  const int chnk = blockIdx.x & 7;     // which group of 8 N-tiles

  const float* xb = x + (size_t)bidx * Cdim * Ndim;

  // ---- one-time preload: Vw [16][384] and U [64][16] into LDS ----
  for (int i = tid; i < Rdim * (Cdim / 4); i += 256) {   // 16*96 float4s
    int r = i / (Cdim / 4), f = i % (Cdim / 4);
    float4 v = *(const float4*)(Vw + (size_t)r * Cdim + 4 * f);
    *(float4*)(sVw + r * VWS + 4 * f) = v;
  }
  {
    int r = tid >> 2, f = tid & 3;                        // 64*4 float4s == 256 threads
    float4 u = *(const float4*)(U + (size_t)r * Rdim + 4 * f);
    *(float4*)(sU + r * US + 4 * f) = u;
  }
  // bias for accumulator init of V (row = v + 8*hf)
  float vb[8];
#pragma unroll
  for (int v = 0; v < 8; ++v) vb[v] = Vb[v + 8 * hf];

  // persistent T accumulators: wave owns c-tiles {wave, wave+8, wave+16} x 4 m-tiles
  v8f accT[3][4];
#pragma unroll
  for (int cg = 0; cg < 3; ++cg)
#pragma unroll
    for (int mt = 0; mt < 4; ++mt)
#pragma unroll
      for (int v = 0; v < 8; ++v) accT[cg][mt][v] = 0.0f;

  // ---- prologue: TDM-prefetch tile 0 into buffer 0 ----
  if (wave == 0) {
    tdm_load_x_tile(xb + chnk * (TILES_PER_WG * NT), OFF_X0 * 4u);
  }

  for (int it = 0; it < TILES_PER_WG; ++it) {
    const int n_base = chnk * (TILES_PER_WG * NT) + it * NT;

    // ---- issue next tile's DMA into the other buffer, wait for current tile
    if (wave == 0) {
      if (it + 1 < TILES_PER_WG) {
        const unsigned nbuf_off = ((it + 1) & 1) ? (unsigned)(OFF_X1 * 4) : 0u;
        tdm_load_x_tile(xb + n_base + NT, nbuf_off);
        __builtin_amdgcn_s_wait_tensorcnt(1);   // tile `it` resident (in-order)
      } else {
        __builtin_amdgcn_s_wait_tensorcnt(0);
      }
    }
    __syncthreads();   // publish tile `it` to all waves

    float* sX = smem + ((it & 1) ? OFF_X1 : OFF_X0);

    // ---- phase 1: V[16][64] = Vw * X + Vb  (waves 0-3, one 16-col subtile each)
    if (wave < 4) {
      const int n0 = wave * 16;
      v8f acc;
#pragma unroll
      for (int v = 0; v < 8; ++v) acc[v] = vb[v];
      for (int k = 0; k < Cdim; k += 4) {
        const int ka = k + 2 * hf;
        v2f a, bfr;
        a[0]   = sVw[l16 * VWS + ka];
        a[1]   = sVw[l16 * VWS + ka + 1];
        bfr[0] = sX[(ka) * XS + n0 + l16];
        bfr[1] = sX[(ka + 1) * XS + n0 + l16];
        acc = wmma_f32(a, bfr, acc);
      }
#pragma unroll
      for (int v = 0; v < 8; ++v) sV[(v + 8 * hf) * VS + n0 + l16] = acc[v];
    }
    __syncthreads();

    // ---- phase 2: L[64][64] = U * V  (waves 0-3) ----
    if (wave < 4) {
      const int n0 = wave * 16;
#pragma unroll
      for (int mt = 0; mt < 4; ++mt) {
        v8f acc;
#pragma unroll
        for (int v = 0; v < 8; ++v) acc[v] = 0.0f;
#pragma unroll
        for (int k = 0; k < Rdim; k += 4) {
          const int ka = k + 2 * hf;
          v2f a, bfr;
          a[0]   = sU[(mt * 16 + l16) * US + ka];
          a[1]   = sU[(mt * 16 + l16) * US + ka + 1];
          bfr[0] = sV[(ka) * VS + n0 + l16];
          bfr[1] = sV[(ka + 1) * VS + n0 + l16];
          acc = wmma_f32(a, bfr, acc);
        }
#pragma unroll
        for (int v = 0; v < 8; ++v)
          sS[(mt * 16 + v + 8 * hf) * SS + n0 + l16] = acc[v];
      }
    }
    __syncthreads();

    // ---- softmax over token dim M (per column; thread owns one column) ----
    if (tid < NT) {
      const int col = tid;
      float mx = -3.402823466e38f;
      for (int m = 0; m < Mdim; ++m) mx = fmaxf(mx, sS[m * SS + col]);
      float sum = 0.0f;
      for (int m = 0; m < Mdim; ++m) {
        float e = __expf(sS[m * SS + col] - mx);
        sum += e;
        sS[m * SS + col] = e;
      }
      const float inv = 1.0f / sum;
      float* so = Sout + (size_t)bidx * Mdim * Ndim + n_base + col;
      for (int m = 0; m < Mdim; ++m) {
        float s = sS[m * SS + col] * inv;
        sS[m * SS + col] = s;
        __builtin_nontemporal_store(s, so + (size_t)m * Ndim);  // streamed output
      }
    }
    __syncthreads();

    // ---- phase 3: T[384][64] += X * S^T  (all 8 waves, persistent accs) ----
    for (int k = 0; k < NT; k += 4) {
      const int ka = k + 2 * hf;
      v2f bfr[4];
#pragma unroll
      for (int mt = 0; mt < 4; ++mt) {
        bfr[mt][0] = sS[(mt * 16 + l16) * SS + ka];
        bfr[mt][1] = sS[(mt * 16 + l16) * SS + ka + 1];
      }
#pragma unroll
      for (int cg = 0; cg < 3; ++cg) {
        const int c0 = (wave + cg * 8) * 16;
        v2f a;
        a[0] = sX[(c0 + l16) * XS + ka];
        a[1] = sX[(c0 + l16) * XS + ka + 1];
#pragma unroll
        for (int mt = 0; mt < 4; ++mt)
          accT[cg][mt] = wmma_f32(a, bfr[mt], accT[cg][mt]);
      }
    }
    __syncthreads();  // buffer consumed; safe target for next TDM issue
  }

  // ---- flush persistent T accumulators with f32 atomics (8x fewer than per-tile)
  float* tb = Tout + (size_t)bidx * Cdim * Mdim;
#pragma unroll
  for (int cg = 0; cg < 3; ++cg) {
    const int c0 = (wave + cg * 8) * 16;
#pragma unroll
    for (int mt = 0; mt < 4; ++mt) {
#pragma unroll
      for (int v = 0; v < 8; ++v) {
        const int c = c0 + v + 8 * hf;
        const int m = mt * 16 + l16;
        unsafeAtomicAdd(tb + (size_t)c * Mdim + m, accT[cg][mt][v]);
      }
    }
  }
}

extern "C" void kernel_launch(void* const* d_in, const int* in_sizes, int n_in,
                              void* d_out, int out_size, void* d_ws, size_t ws_size,
                              hipStream_t stream) {
  const float* x  = (const float*)d_in[0];
  const float* Vw = (const float*)d_in[1];
  const float* Vb = (const float*)d_in[2];
  const float* U  = (const float*)d_in[3];

  float* Tout = (float*)d_out;                                // [B,C,M]
  float* Sout = Tout + (size_t)Bsz * Cdim * Mdim;             // [B,M,N]

  const int tN = Bsz * Cdim * Mdim;                           // 786432
  zero_f32_kernel<<<(tN + 255) / 256, 256, 0, stream>>>(Tout, tN);

  const size_t lds_bytes = (size_t)LDS_FLOATS * sizeof(float); // 260480
  tokenpool_fused_kernel<<<dim3(Bsz * WGS_PER_BATCH), dim3(256), lds_bytes,
                           stream>>>(x, Vw, Vb, U, Tout, Sout);
}